// Rroi_align_61014305407554
// MI455X (gfx1250) — compile-verified
//
#include <hip/hip_runtime.h>
#include <hip/hip_bf16.h>
#include <stdint.h>

// Rotated ROI Align for MI455X (gfx1250).
// Shapes fixed by reference setup: features (B=4, W=256, H=256, C=256) f32,
// rois (R, 6) f32, pooled 7x7.  Gather-bandwidth-bound (~129 MB -> ~5.5 us at
// 23.3 TB/s).  No matrix structure exists (the length-4 contraction operand
// depends on the output row, so WMMA cannot express it); the CDNA5 path used
// is the ASYNCcnt-tracked async global->LDS copy for the 4 corner-row
// gathers, with a single s_wait_asynccnt before the blend.  Per-bin transform
// math is uniform per block and scalarizes onto the SALU.

#define FB 4
#define FW 256
#define FH 256
#define FC 256
#define PH 7
#define PW 7
#define PIF 3.1415926535f

__global__ __launch_bounds__(64) void rroi_align_kernel(
    const float* __restrict__ features,
    const float* __restrict__ rois,
    const float* __restrict__ ss_ptr,
    float* __restrict__ out,        // (R, PH, PW, FC)
    float* __restrict__ conx,       // (R, PH, PW)
    float* __restrict__ cony,       // (R, PH, PW)
    int R)
{
    const int bin = blockIdx.x;                 // r*PH*PW + i*PW + j
    if (bin >= R * PH * PW) return;
    const int j = bin % PW;
    const int i = (bin / PW) % PH;
    const int r = bin / (PW * PH);
    const int t = threadIdx.x;                  // 0..63 -> channels 4t..4t+3

    // ---- per-bin transform (uniform across block -> SALU) ----
    const float ss = ss_ptr[0];
    const float* roi = rois + r * 6;
    const int   b     = (int)roi[0];
    const float cx    = roi[1];
    const float cy    = roi[2];
    const float rh    = roi[3];
    const float rw    = roi[4];
    const float theta = roi[5] * 180.0f * PIF;

    const float roi_pw = rw / rh * (float)PW;
    const float dxv = -roi_pw * 0.5f;
    const float dyv = -(float)PH * 0.5f;
    const float Sx  = rw / roi_pw * ss;
    const float Sy  = rh / (float)PH * ss;
    const float al  = cosf(theta);
    const float be  = sinf(theta);
    const float Dx  = cx * ss;
    const float Dy  = cy * ss;
    const float M00 = al * Sx;
    const float M01 = be * Sy;
    const float M02 = M00 * dxv + M01 * dyv + Dx;
    const float M10 = -be * Sx;
    const float M11 = al * Sy;
    const float M12 = M10 * dxv + M11 * dyv + Dy;

    // linspace(0, PH, PH): step = PH/(PH-1)
    const float xv = (float)i * ((float)PH / (float)(PH - 1));
    const float yv = (float)j * ((float)PW / (float)(PW - 1));

    const float P0 = M00 * xv          + M01 * yv          + M02;
    const float P1 = M10 * xv          + M11 * yv          + M12;
    const float P2 = M00 * xv          + M01 * (yv + 1.0f) + M02;
    const float P3 = M10 * xv          + M11 * (yv + 1.0f) + M12;
    const float P4 = M00 * (xv + 1.0f) + M01 * yv          + M02;
    const float P5 = M10 * (xv + 1.0f) + M11 * yv          + M12;
    const float P6 = M00 * (xv + 1.0f) + M01 * (yv + 1.0f) + M02;
    const float P7 = M10 * (xv + 1.0f) + M11 * (yv + 1.0f) + M12;

    // rintf == jnp.round (round-half-even under default RTE)
    const float leftMost   = fmaxf(rintf(fminf(fminf(P0, P2), fminf(P4, P6))), 0.0f);
    const float rightMost  = fminf(rintf(fmaxf(fmaxf(P0, P2), fmaxf(P4, P6))), (float)FW - 1.0f);
    const float topMost    = fmaxf(rintf(fminf(fminf(P1, P3), fminf(P5, P7))), 0.0f);
    const float bottomMost = fminf(rintf(fmaxf(fmaxf(P1, P3), fmaxf(P5, P7))), (float)FH - 1.0f);

    const float bin_cx = (leftMost + rightMost) * 0.5f;
    const float bin_cy = (topMost + bottomMost) * 0.5f;

    const float bl = floorf(bin_cx);
    const float br = ceilf(bin_cx);
    const float bt = floorf(bin_cy);
    const float bb = ceilf(bin_cy);

    const bool vl = (bl >= 0.0f) && (bl <= (float)FW - 1.0f);
    const bool vr = (br >= 0.0f) && (br <= (float)FW - 1.0f);
    const bool vt = (bt >= 0.0f) && (bt <= (float)FH - 1.0f);
    const bool vb = (bb >= 0.0f) && (bb <= (float)FH - 1.0f);

    const int li = (int)fminf(fmaxf(bl, 0.0f), (float)FW - 1.0f);
    const int ri = (int)fminf(fmaxf(br, 0.0f), (float)FW - 1.0f);
    const int ti = (int)fminf(fmaxf(bt, 0.0f), (float)FH - 1.0f);
    const int bi = (int)fminf(fmaxf(bb, 0.0f), (float)FH - 1.0f);

    const float rx = bin_cx - bl;
    const float ry = bin_cy - bt;
    // fold validity masks into bilinear weights (matches vals*mask semantics)
    const float wlt = (1.0f - rx) * (1.0f - ry) * (float)(vl && vt);
    const float wrt = rx * (1.0f - ry)          * (float)(vr && vt);
    const float wrb = rx * ry                   * (float)(vr && vb);
    const float wlb = (1.0f - rx) * ry          * (float)(vl && vb);

    // ---- async gather of the 4 corner rows into LDS (CDNA5 ASYNC path) ----
    // corner order in LDS: 0=lt(li,ti) 1=rt(ri,ti) 2=rb(ri,bi) 3=lb(li,bi)
    __shared__ __align__(16) float smem[4 * FC];        // 4 KB
    const uint32_t lds0 = (uint32_t)(uintptr_t)(&smem[0]);
    const uint32_t ldsB = lds0 + (uint32_t)(4 * t) * 4u; // this lane's 16B slot

    const size_t ch = (size_t)(4 * t);
    const float* pLT = features + (((size_t)((b * FW + li) * FH + ti)) * FC) + ch;
    const float* pRT = features + (((size_t)((b * FW + ri) * FH + ti)) * FC) + ch;
    const float* pRB = features + (((size_t)((b * FW + ri) * FH + bi)) * FC) + ch;
    const float* pLB = features + (((size_t)((b * FW + li) * FH + bi)) * FC) + ch;

    asm volatile("global_load_async_to_lds_b128 %0, %1, off"
                 :: "v"(ldsB + 0u * FC * 4u), "v"((uint64_t)(uintptr_t)pLT) : "memory");
    asm volatile("global_load_async_to_lds_b128 %0, %1, off"
                 :: "v"(ldsB + 1u * FC * 4u), "v"((uint64_t)(uintptr_t)pRT) : "memory");
    asm volatile("global_load_async_to_lds_b128 %0, %1, off"
                 :: "v"(ldsB + 2u * FC * 4u), "v"((uint64_t)(uintptr_t)pRB) : "memory");
    asm volatile("global_load_async_to_lds_b128 %0, %1, off"
                 :: "v"(ldsB + 3u * FC * 4u), "v"((uint64_t)(uintptr_t)pLB) : "memory");
    asm volatile("s_wait_asynccnt 0" ::: "memory");
    // each lane reads back only the bytes it wrote -> no barrier needed

    const float4 flt = *(const float4*)&smem[0 * FC + 4 * t];
    const float4 frt = *(const float4*)&smem[1 * FC + 4 * t];
    const float4 frb = *(const float4*)&smem[2 * FC + 4 * t];
    const float4 flb = *(const float4*)&smem[3 * FC + 4 * t];

    float4 o;
    o.x = flt.x * wlt + frt.x * wrt + frb.x * wrb + flb.x * wlb;
    o.y = flt.y * wlt + frt.y * wrt + frb.y * wrb + flb.y * wlb;
    o.z = flt.z * wlt + frt.z * wrt + frb.z * wrb + flb.z * wlb;
    o.w = flt.w * wlt + frt.w * wrt + frb.w * wrb + flb.w * wlb;

    *(float4*)(out + (size_t)bin * FC + ch) = o;

    if (t == 0) {
        conx[bin] = bl;   // floor(bin_cx), unclipped, unmasked (matches ref)
        cony[bin] = bt;   // floor(bin_cy)
    }
}

extern "C" void kernel_launch(void* const* d_in, const int* in_sizes, int n_in,
                              void* d_out, int out_size, void* d_ws, size_t ws_size,
                              hipStream_t stream) {
    const float* features = (const float*)d_in[0];
    const float* rois     = (const float*)d_in[1];
    const float* ss       = (const float*)d_in[2];
    // d_in[3], d_in[4]: pooled_height/width == 7 (fixed by reference setup)

    const int R    = in_sizes[1] / 6;
    const int bins = R * PH * PW;

    float* out  = (float*)d_out;
    float* conx = out + (size_t)bins * FC;
    float* cony = conx + (size_t)bins;

    rroi_align_kernel<<<bins, 64, 0, stream>>>(features, rois, ss, out, conx, cony, R);
}